// MPS_7490422964935
// MI455X (gfx1250) — compile-verified
//
#include <hip/hip_runtime.h>

// Problem constants (from reference setup_inputs)
constexpr int kL = 48;   // sites
constexpr int kd = 2;    // physical dim
constexpr int kD = 64;   // bond dim
constexpr int kN = 512;  // batch
constexpr int kNB = 4;           // blocks for amplitude chain
constexpr int kRows = kN / kNB;  // 128 samples per block

typedef __attribute__((ext_vector_type(2))) float v2f;
typedef __attribute__((ext_vector_type(8))) float v8f;

// --- CDNA5 fp32 WMMA: D(16x16) += A(16x4) * B(4x16) ---
__device__ __forceinline__ v8f wmma4(v2f a, v2f b, v8f c) {
  return __builtin_amdgcn_wmma_f32_16x16x4_f32(
      /*neg_a=*/false, a, /*neg_b=*/false, b,
      /*c_mod=*/(short)0, c, /*reuse_a=*/false, /*reuse_b=*/false);
}

// A fragment from row-major M x K source: element j = A[lane&15][kb + 2*(lane>>4) + j]
__device__ __forceinline__ v2f ldA(const float* base, int ld, int lane, int kb) {
  const float* p = base + (lane & 15) * ld + kb + 2 * (lane >> 4);
  v2f r; r.x = p[0]; r.y = p[1];
  return r;
}
// B fragment from row-major K x N source: element j = B[kb + 2*(lane>>4) + j][lane&15]
// (also feeds a transposed matrix into the A operand: A = S^T)
__device__ __forceinline__ v2f ldB(const float* base, int ld, int lane, int kb) {
  const float* p = base + (kb + 2 * (lane >> 4)) * ld + (lane & 15);
  v2f r; r.x = p[0]; r.y = p[ld];
  return r;
}

// =====================================================================
// Amplitude chain: one launch, 4 blocks x 128 samples.
// v ping-pongs between two LDS buffers; one barrier per site.
// Per site: v_out = s0 (*) (v @ B0) + s1 (*) (v @ B1)   (all complex)
// =====================================================================
__global__ void amp_chain(const int* __restrict__ x,
                          const float* __restrict__ rot_r, const float* __restrict__ rot_i,
                          const float* __restrict__ left_r, const float* __restrict__ left_i,
                          const float* __restrict__ bulk_r, const float* __restrict__ bulk_i,
                          const float* __restrict__ right_r, const float* __restrict__ right_i,
                          float* __restrict__ logp) {
  extern __shared__ float smem[];
  float* vrA = smem;                      // kRows*kD each
  float* viA = vrA + kRows * kD;
  float* vrB = viA + kRows * kD;
  float* viB = vrB + kRows * kD;
  const int tid = threadIdx.x;
  const int lane = tid & 31, wid = tid >> 5;
  const int n0 = blockIdx.x * kRows;

  // ---- init: v[n,j] = sum_t s_t(n, l=0) * left[t,0,j] ----
  for (int idx = tid; idx < kRows * kD; idx += blockDim.x) {
    int n = n0 + (idx >> 6), j = idx & 63;
    int xi = x[n * kL];
    int base = n * kL * (kd * kd) + xi * kd;
    float s0r = rot_r[base], s1r = rot_r[base + 1];
    float s0i = rot_i[base], s1i = rot_i[base + 1];
    float l0r = left_r[j], l1r = left_r[kD + j];
    float l0i = left_i[j], l1i = left_i[kD + j];
    vrA[idx] = s0r * l0r - s0i * l0i + s1r * l1r - s1i * l1i;
    viA[idx] = s0r * l0i + s0i * l0r + s1r * l1i + s1i * l1r;
  }
  __syncthreads();

  // ---- 46 fused site steps; 32 tiles (8 tm x 4 tn) over 8 waves ----
#pragma unroll 1
  for (int ls = 0; ls < kL - 2; ++ls) {
    const float* src_r = (ls & 1) ? vrB : vrA;
    const float* src_i = (ls & 1) ? viB : viA;
    float* dst_r = (ls & 1) ? vrA : vrB;
    float* dst_i = (ls & 1) ? viA : viB;
    const float* Br = bulk_r + (size_t)ls * kd * kD * kD;
    const float* Bi = bulk_i + (size_t)ls * kd * kD * kD;
    const int l = ls + 1;
#pragma unroll 1
    for (int q = 0; q < 4; ++q) {
      int tile = wid * 4 + q;          // 0..31
      int tm = tile >> 2, tn = tile & 3;
      const float* A_r = src_r + tm * 16 * kD;   // LDS
      const float* A_i = src_i + tm * 16 * kD;   // LDS
      const float* B0r = Br + tn * 16;
      const float* B0i = Bi + tn * 16;
      const float* B1r = Br + kD * kD + tn * 16;
      const float* B1i = Bi + kD * kD + tn * 16;
      v8f RR0 = {}, II0 = {}, RI0 = {}, IR0 = {};
      v8f RR1 = {}, II1 = {}, RI1 = {}, IR1 = {};
#pragma unroll
      for (int kb = 0; kb < kD; kb += 4) {
        v2f ar = ldA(A_r, kD, lane, kb);
        v2f ai = ldA(A_i, kD, lane, kb);
        v2f b0r = ldB(B0r, kD, lane, kb);
        v2f b0i = ldB(B0i, kD, lane, kb);
        v2f b1r = ldB(B1r, kD, lane, kb);
        v2f b1i = ldB(B1i, kD, lane, kb);
        RR0 = wmma4(ar, b0r, RR0);
        II0 = wmma4(ai, b0i, II0);
        RI0 = wmma4(ar, b0i, RI0);
        IR0 = wmma4(ai, b0r, IR0);
        RR1 = wmma4(ar, b1r, RR1);
        II1 = wmma4(ai, b1i, II1);
        RI1 = wmma4(ar, b1i, RI1);
        IR1 = wmma4(ai, b1r, IR1);
      }
      // combine with per-sample rotation scalars, store straight to dst LDS
      int h = lane >> 4, c = lane & 15;
#pragma unroll
      for (int e = 0; e < 8; ++e) {
        int row = tm * 16 + e + 8 * h;        // local sample row
        int n = n0 + row;                      // global sample
        int xi = x[n * kL + l];
        int base = (n * kL + l) * (kd * kd) + xi * kd;
        float s0r = rot_r[base], s1r = rot_r[base + 1];
        float s0i = rot_i[base], s1i = rot_i[base + 1];
        float p0r = RR0[e] - II0[e], p0i = RI0[e] + IR0[e];
        float p1r = RR1[e] - II1[e], p1i = RI1[e] + IR1[e];
        dst_r[row * kD + tn * 16 + c] = s0r * p0r - s0i * p0i + s1r * p1r - s1i * p1i;
        dst_i[row * kD + tn * 16 + c] = s0r * p0i + s0i * p0r + s1r * p1i + s1i * p1r;
      }
    }
    // prefetch next site's matrices into cache (global_prefetch_b8)
    if (ls + 1 < kL - 2) {
      const float* pr = bulk_r + (size_t)(ls + 1) * kd * kD * kD;
      const float* pi = bulk_i + (size_t)(ls + 1) * kd * kD * kD;
      __builtin_prefetch(pr + tid * 32, 0, 1);  // 256 threads x 128B = 32KB
      __builtin_prefetch(pi + tid * 32, 0, 1);
    }
    __syncthreads();  // dst fully written, src fully consumed
  }

  // 46 steps (even) -> final v in the A buffers
  // ---- epilogue: logp[n] = log |v . (s0*right0 + s1*right1)|^2 ----
  if (tid < kRows) {
    int n = n0 + tid;
    int l = kL - 1;
    int xi = x[n * kL + l];
    int base = (n * kL + l) * (kd * kd) + xi * kd;
    float s0r = rot_r[base], s1r = rot_r[base + 1];
    float s0i = rot_i[base], s1i = rot_i[base + 1];
    float ar = 0.f, ai = 0.f;
    for (int j = 0; j < kD; ++j) {
      float rr0 = right_r[j], ri0 = right_i[j];
      float rr1 = right_r[kD + j], ri1 = right_i[kD + j];
      float Rr = s0r * rr0 - s0i * ri0 + s1r * rr1 - s1i * ri1;
      float Ri = s0r * ri0 + s0i * rr0 + s1r * ri1 + s1i * rr1;
      float wr = vrA[tid * kD + j], wi = viA[tid * kD + j];
      ar += wr * Rr - wi * Ri;
      ai += wr * Ri + wi * Rr;
    }
    logp[n] = logf(ar * ar + ai * ai);
  }
}

// =====================================================================
// Norm chain: one launch, one block. c and T live in LDS.
// Per site: T_s = S_s^T @ c ;  c' = sum_s T_s @ conj(S_s). Then epilogue
// (right-boundary contraction, log, and the final NLL combine).
// =====================================================================
__global__ void norm_chain(const float* __restrict__ left_r, const float* __restrict__ left_i,
                           const float* __restrict__ bulk_r, const float* __restrict__ bulk_i,
                           const float* __restrict__ right_r, const float* __restrict__ right_i,
                           const float* __restrict__ logp, float* __restrict__ out) {
  extern __shared__ float smem[];
  float* cr = smem;                  // kD*kD
  float* ci = cr + kD * kD;          // kD*kD
  float* Tr = ci + kD * kD;          // kd*kD*kD
  float* Ti = Tr + kd * kD * kD;     // kd*kD*kD
  float* red = Ti + kd * kD * kD;    // 256 + 1
  const int tid = threadIdx.x;
  const int lane = tid & 31, wid = tid >> 5;

  // ---- init c from left boundary ----
  for (int idx = tid; idx < kD * kD; idx += blockDim.x) {
    int j = idx >> 6, k = idx & 63;
    float acc_r = 0.f, acc_i = 0.f;
    for (int s = 0; s < kd; ++s) {
      float ajr = left_r[s * kD + j], aji = left_i[s * kD + j];
      float akr = left_r[s * kD + k], aki = left_i[s * kD + k];
      acc_r += ajr * akr + aji * aki;
      acc_i += aji * akr - ajr * aki;
    }
    cr[idx] = acc_r;
    ci[idx] = acc_i;
  }
  __syncthreads();

#pragma unroll 1
  for (int ls = 0; ls < kL - 2; ++ls) {
    const float* Sr = bulk_r + (size_t)ls * kd * kD * kD;
    const float* Si = bulk_i + (size_t)ls * kd * kD * kD;

    // ---- stage1: T_s = S_s^T @ c ; 32 tiles, 4 per wave ----
#pragma unroll 1
    for (int q = 0; q < 4; ++q) {
      int tile = wid * 4 + q;          // 0..31
      int s = tile >> 4, t2 = tile & 15;
      int tm = t2 >> 2, tn = t2 & 3;
      const float* sr = Sr + s * kD * kD + tm * 16;  // A = S^T via ldB access
      const float* si = Si + s * kD * kD + tm * 16;
      const float* br = cr + tn * 16;  // LDS
      const float* bi = ci + tn * 16;  // LDS
      v8f RR = {}, II = {}, RI = {}, IR = {};
#pragma unroll
      for (int kb = 0; kb < kD; kb += 4) {
        v2f atr = ldB(sr, kD, lane, kb);
        v2f ati = ldB(si, kD, lane, kb);
        v2f bcr = ldB(br, kD, lane, kb);
        v2f bci = ldB(bi, kD, lane, kb);
        RR = wmma4(atr, bcr, RR);
        II = wmma4(ati, bci, II);
        RI = wmma4(atr, bci, RI);
        IR = wmma4(ati, bcr, IR);
      }
      int h = lane >> 4, c = lane & 15;
#pragma unroll
      for (int e = 0; e < 8; ++e) {
        int row = tm * 16 + e + 8 * h;
        Tr[s * kD * kD + row * kD + tn * 16 + c] = RR[e] - II[e];
        Ti[s * kD * kD + row * kD + tn * 16 + c] = RI[e] + IR[e];
      }
    }
    __syncthreads();

    // ---- stage2: c = sum_s T_s @ conj(S_s) ; 16 tiles, 2 per wave ----
#pragma unroll 1
    for (int q = 0; q < 2; ++q) {
      int tile = wid * 2 + q;          // 0..15
      int tm = tile >> 2, tn = tile & 3;
      v8f ARR = {}, AII = {}, AIR = {}, ARI = {};
#pragma unroll
      for (int s = 0; s < kd; ++s) {
        const float* ar_ = Tr + s * kD * kD + tm * 16 * kD;  // LDS
        const float* ai_ = Ti + s * kD * kD + tm * 16 * kD;  // LDS
        const float* br_ = Sr + s * kD * kD + tn * 16;
        const float* bi_ = Si + s * kD * kD + tn * 16;
#pragma unroll
        for (int kb = 0; kb < kD; kb += 4) {
          v2f ar = ldA(ar_, kD, lane, kb);
          v2f ai = ldA(ai_, kD, lane, kb);
          v2f br = ldB(br_, kD, lane, kb);
          v2f bi = ldB(bi_, kD, lane, kb);
          ARR = wmma4(ar, br, ARR);
          AII = wmma4(ai, bi, AII);
          AIR = wmma4(ai, br, AIR);
          ARI = wmma4(ar, bi, ARI);
        }
      }
      int h = lane >> 4, c = lane & 15;
#pragma unroll
      for (int e = 0; e < 8; ++e) {
        int row = tm * 16 + e + 8 * h;
        cr[row * kD + tn * 16 + c] = ARR[e] + AII[e];
        ci[row * kD + tn * 16 + c] = AIR[e] - ARI[e];
      }
    }
    // prefetch next site's matrices
    if (ls + 1 < kL - 2) {
      const float* pr = bulk_r + (size_t)(ls + 1) * kd * kD * kD;
      const float* pi = bulk_i + (size_t)(ls + 1) * kd * kD * kD;
      __builtin_prefetch(pr + tid * 32, 0, 1);
      __builtin_prefetch(pi + tid * 32, 0, 1);
    }
    __syncthreads();
  }

  // ---- norm = Re( sum_s right_s^T c conj(right_s) ) ----
  float acc = 0.f;
  for (int idx = tid; idx < kd * kD * kD; idx += blockDim.x) {
    int s = idx >> 12, rem = idx & 4095;
    int i = rem >> 6, j = rem & 63;
    float a = right_r[s * kD + i], b = right_i[s * kD + i];
    float e = right_r[s * kD + j], f = right_i[s * kD + j];
    float cre = cr[i * kD + j], cim = ci[i * kD + j];
    float pr = a * cre - b * cim;
    float pi = a * cim + b * cre;
    acc += pr * e + pi * f;  // Re( p * conj(r_j) )
  }
  red[tid] = acc;
  __syncthreads();
  for (int off = 128; off > 0; off >>= 1) {
    if (tid < (unsigned)off) red[tid] += red[tid + off];
    __syncthreads();
  }
  if (tid == 0) red[256] = red[0];  // save norm
  __syncthreads();

  // ---- reduce logp and emit the final scalar ----
  red[tid] = logp[tid] + logp[tid + 256];
  __syncthreads();
  for (int off = 128; off > 0; off >>= 1) {
    if (tid < (unsigned)off) red[tid] += red[tid + off];
    __syncthreads();
  }
  if (tid == 0) out[0] = -(red[0] / (float)kN) + logf(red[256]);
}

extern "C" void kernel_launch(void* const* d_in, const int* in_sizes, int n_in,
                              void* d_out, int out_size, void* d_ws, size_t ws_size,
                              hipStream_t stream) {
  const int*   x       = (const int*)  d_in[0];
  const float* rot_r   = (const float*)d_in[1];
  const float* rot_i   = (const float*)d_in[2];
  const float* left_r  = (const float*)d_in[3];
  const float* left_i  = (const float*)d_in[4];
  const float* bulk_r  = (const float*)d_in[5];
  const float* bulk_i  = (const float*)d_in[6];
  const float* right_r = (const float*)d_in[7];
  const float* right_i = (const float*)d_in[8];
  float* out  = (float*)d_out;
  float* logp = (float*)d_ws;  // kN floats

  const size_t amp_lds  = (size_t)(4 * kRows * kD) * sizeof(float);  // 128 KB (ping-pong)
  const size_t norm_lds = (size_t)(2 * kD * kD + 2 * kd * kD * kD + 257) * sizeof(float);

  amp_chain<<<kNB, 256, amp_lds, stream>>>(x, rot_r, rot_i, left_r, left_i,
                                           bulk_r, bulk_i, right_r, right_i, logp);
  norm_chain<<<1, 256, norm_lds, stream>>>(left_r, left_i, bulk_r, bulk_i,
                                           right_r, right_i, logp, out);
}